// LinearPolynomialCell_16801912062690
// MI455X (gfx1250) — compile-verified
//
#include <hip/hip_runtime.h>
#include <cmath>

// Problem sizes (fixed by the reference)
#define DIMD   1024
#define BATCH  16
#define TSTEPS 2048
#define MROWS  (TSTEPS * BATCH)   // 32768 rows in the flattened [T*B, D] GEMMs
#define TBD    (TSTEPS * BATCH * DIMD)

typedef __attribute__((ext_vector_type(2))) float v2f;
typedef __attribute__((ext_vector_type(4))) float v4f;
typedef __attribute__((ext_vector_type(8))) float v8f;

// ---------------------------------------------------------------------------
// WMMA: D = A(16x4 f32) * B(4x16 f32) + C(16x16 f32), wave32. (codegen-OK R1)
// ---------------------------------------------------------------------------
static __device__ inline v8f wmma_f32_k4(v2f a, v2f b, v8f c) {
#if defined(__gfx1250__) && __has_builtin(__builtin_amdgcn_wmma_f32_16x16x4_f32)
  return __builtin_amdgcn_wmma_f32_16x16x4_f32(
      /*neg_a=*/false, a, /*neg_b=*/false, b,
      /*c_mod=*/(short)0, c, /*reuse_a=*/false, /*reuse_b=*/false);
#else
  c[0] += a.x * b.x + a.y * b.y;   // host-pass / missing-builtin placeholder
  return c;
#endif
}

// ---------------------------------------------------------------------------
// Async global -> LDS 16B block copy (CDNA5 GLOBAL_LOAD_ASYNC_TO_LDS_B128,
// tracked by ASYNCcnt). LDS operand is the low 32 bits of the generic shared
// pointer (ISA 10.2: LDS aperture addresses truncate to addr[31:0]).
// ---------------------------------------------------------------------------
static __device__ inline void async_ld16(void* lds, const void* g) {
#if defined(__gfx1250__)
  unsigned off = (unsigned)(size_t)lds;
  asm volatile("global_load_async_to_lds_b128 %0, %1, off"
               :: "v"(off), "v"(g) : "memory");
#else
  *(v4f*)lds = *(const v4f*)g;     // host-pass placeholder
#endif
}

static __device__ inline void wait_async0() {
#if defined(__gfx1250__)
  asm volatile("s_wait_asynccnt 0" ::: "memory");
#endif
}

// ---------------------------------------------------------------------------
// Fused projection GEMM:
//   vx    = X @ Wx^T + bx
//   alpha = 1 + softplus(X @ Wa^T + ba)
//   delta = sigmoid(X @ Wd^T + bd)
// X: [MROWS, D] row-major, W*: [e, d] row-major (einsum tbd,ed->tbe).
// Block: 256 threads (8 wave32). Tile: 128(M) x 64(N), K-chunk 32.
// A and B tiles staged ROW-MAJOR in LDS (stride 36 floats):
//   - every global->LDS transfer is a pure 16B block  -> async b128 copies
//   - A frag  = v2f at As[m][k0]      (single aligned ds_load_b64)
//   - B frag  = v2f at Bs[n][k0]      (B[k][n] = W[n][k], contiguous in k)
//   - stride 36: 32-lane fragment reads cover all 64 LDS banks exactly once
// ---------------------------------------------------------------------------
#define BM  128
#define BN  64
#define KC  32
#define LDT 36   // padded LDS row stride (floats) for both tiles

__global__ __launch_bounds__(256)
void fused_proj_kernel(const float* __restrict__ X,
                       const float* __restrict__ Wx,
                       const float* __restrict__ Wa,
                       const float* __restrict__ Wd,
                       const float* __restrict__ bx,
                       const float* __restrict__ ba,
                       const float* __restrict__ bd,
                       float* __restrict__ vx_out,
                       float* __restrict__ alpha_out,
                       float* __restrict__ delta_out) {
  __shared__ float As[BM * LDT];          // 18.0 KB
  __shared__ float Bs[3][BN * LDT];       // 3 x 9.0 KB

  const int tid     = threadIdx.x;
  const int wave    = tid >> 5;
  const int lane    = tid & 31;
  const int rowBase = blockIdx.x * BM;
  const int colBase = blockIdx.y * BN;

  const float* const Ws[3] = {Wx, Wa, Wd};

  v8f acc[3][4];
#pragma unroll
  for (int w = 0; w < 3; ++w)
#pragma unroll
    for (int ns = 0; ns < 4; ++ns)
      acc[w][ns] = (v8f){0.f, 0.f, 0.f, 0.f, 0.f, 0.f, 0.f, 0.f};

  // Per-lane fragment coordinates (ISA 7.12.2, wave32):
  //  A 16x4 f32:  VGPR j, lane l -> M = l%16, K = j + 2*(l/16)
  //  B  4x16 f32: VGPR j, lane l -> N = l%16, K = j + 2*(l/16)
  //  C 16x16 f32: VGPR i, lane l -> M = i + 8*(l/16), N = l%16
  const int mloc  = wave * 16 + (lane & 15);
  const int khalf = (lane >> 4) * 2;
  const int nlane = lane & 15;

  for (int kc = 0; kc < DIMD; kc += KC) {
    // ---- Stage A tile: 128 x 32 floats = 1024 x 16B chunks, 4/thread ----
#pragma unroll
    for (int i = 0; i < 4; ++i) {
      int idx4 = tid + 256 * i;      // 0..1023
      int m    = idx4 >> 3;
      int c    = idx4 & 7;
      async_ld16(&As[m * LDT + 4 * c],
                 X + (size_t)(rowBase + m) * DIMD + kc + 4 * c);
    }
    // ---- Stage B tiles (3 weights), row-major: 64 x 32 floats each ----
#pragma unroll
    for (int w = 0; w < 3; ++w) {
#pragma unroll
      for (int i = 0; i < 2; ++i) {
        int idx4 = tid + 256 * i;    // 0..511
        int n    = idx4 >> 3;        // output column e within tile
        int c    = idx4 & 7;
        async_ld16(&Bs[w][n * LDT + 4 * c],
                   Ws[w] + (size_t)(colBase + n) * DIMD + kc + 4 * c);
      }
    }
    wait_async0();
    __syncthreads();

    // ---- 8 K-steps of 4, full-precision f32 WMMA ----
#pragma unroll
    for (int s = 0; s < 8; ++s) {
      int k0 = 4 * s + khalf;
      v2f a = *(const v2f*)(&As[mloc * LDT + k0]);
#pragma unroll
      for (int w = 0; w < 3; ++w) {
#pragma unroll
        for (int ns = 0; ns < 4; ++ns) {
          v2f b = *(const v2f*)(&Bs[w][(ns * 16 + nlane) * LDT + k0]);
          acc[w][ns] = wmma_f32_k4(a, b, acc[w][ns]);
        }
      }
    }
    __syncthreads();
  }

  // ---- Epilogue: biases + activations, store to workspace ----
  const int rhalf = (lane >> 4) * 8;
#pragma unroll
  for (int ns = 0; ns < 4; ++ns) {
    int e = colBase + ns * 16 + nlane;
    float bxv = bx[e];
    float bav = ba[e];
    float bdv = bd[e];
#pragma unroll
    for (int vi = 0; vi < 8; ++vi) {
      size_t row = (size_t)(rowBase + wave * 16 + rhalf + vi);
      size_t g   = row * DIMD + e;
      vx_out[g] = acc[0][ns][vi] + bxv;
      // alpha = 1 + softplus(z), stable: max(z,0) + log1p(exp(-|z|))
      float za = acc[1][ns][vi] + bav;
      float sp = fmaxf(za, 0.f) + log1pf(expf(-fabsf(za)));
      alpha_out[g] = 1.0f + sp;
      // delta = sigmoid(z)
      float zd = acc[2][ns][vi] + bdv;
      delta_out[g] = 1.0f / (1.0f + expf(-zd));
    }
  }
}

// ---------------------------------------------------------------------------
// Sequential scan: one recurrence chain per (b, d) element, ILP=4 chains per
// thread to fill pipeline bubbles in the dependent log2->exp2->exp chain.
// 4096 threads cover all 16384 chains; loads coalesced along d.
// ---------------------------------------------------------------------------
#define ILP 4

__global__ __launch_bounds__(256)
void scan_kernel(const float* __restrict__ X,       // [T,B,D]
                 const float* __restrict__ h0,      // [B,D]
                 const float* __restrict__ r_h,     // [D]
                 const float* __restrict__ b_gate,  // [D]
                 const float* __restrict__ vx,      // [T,B,D]
                 const float* __restrict__ alpha,   // [T,B,D]
                 const float* __restrict__ delta,   // [T,B,D]
                 float* __restrict__ out,           // [T,B,D]
                 float* __restrict__ hout) {        // [T+1,B,D]
  const int tid = blockIdx.x * blockDim.x + threadIdx.x;  // 0..4095
  const int d   = tid & (DIMD - 1);
  const int q   = tid >> 10;                              // 0..3

  const float rh = r_h[d];
  const float bg = b_gate[d];

  float h[ILP];
  int   idx[ILP];
#pragma unroll
  for (int j = 0; j < ILP; ++j) {
    int b  = q + 4 * j;                  // covers b = 0..15 across (q, j)
    idx[j] = b * DIMD + d;
    h[j]   = h0[idx[j]];
    hout[idx[j]] = h[j];                 // h[0] slice
  }

  for (int t = 0; t < TSTEPS; ++t) {
    const size_t base = (size_t)t * (BATCH * DIMD);
    // Prefetch next timestep's streams while this step's chain executes.
    if (t + 1 < TSTEPS) {
      size_t nb = base + (size_t)(BATCH * DIMD) + idx[0];
      __builtin_prefetch(vx + nb, 0, 0);
      __builtin_prefetch(alpha + nb, 0, 0);
      __builtin_prefetch(delta + nb, 0, 0);
      __builtin_prefetch(X + nb, 0, 0);
    }
#pragma unroll
    for (int j = 0; j < ILP; ++j) {
      size_t g  = base + idx[j];
      float  v  = vx[g] + rh * h[j];
      float  al = alpha[g];
      float  de = delta[g];
      float  xv = X[g];

      float av = fabsf(v);
      av = fminf(fmaxf(av, 1e-6f), 10.0f);
      float mag = exp2f(al * log2f(av));                  // |v|^alpha
      float sgn = (v > 0.f) ? 1.f : ((v < 0.f) ? -1.f : 0.f);
      float cand = sgn * mag;

      h[j] = h[j] + de * (cand - h[j]);                   // (1-d)h + d*cand

      float gt  = h[j] + xv + bg;
      float sig = 1.0f / (1.0f + expf(-gt));
      out[g] = h[j] * (gt * sig);                          // h * silu(gate)
      hout[base + (size_t)(BATCH * DIMD) + idx[j]] = h[j]; // h[t+1]
    }
  }
}

// ---------------------------------------------------------------------------
// Launch
// Inputs (setup_inputs order):
//  0 x [T,B,D]  1 h0 [B,D]  2 W_x [D,D]  3 r_h [D]  4 b [D]
//  5 W_alpha [D,D]  6 b_alpha [D]  7 W_delta [D,D]  8 b_delta [D]  9 b_gate [D]
// Output: concat(output [T,B,D], h [T+1,B,D]) as f32.
// Workspace: vx | alpha | delta, each T*B*D f32 (total ~403 MB).
// ---------------------------------------------------------------------------
extern "C" void kernel_launch(void* const* d_in, const int* in_sizes, int n_in,
                              void* d_out, int out_size, void* d_ws, size_t ws_size,
                              hipStream_t stream) {
  (void)in_sizes; (void)n_in; (void)out_size; (void)ws_size;

  const float* x       = (const float*)d_in[0];
  const float* h0      = (const float*)d_in[1];
  const float* W_x     = (const float*)d_in[2];
  const float* r_h     = (const float*)d_in[3];
  const float* b       = (const float*)d_in[4];
  const float* W_alpha = (const float*)d_in[5];
  const float* b_alpha = (const float*)d_in[6];
  const float* W_delta = (const float*)d_in[7];
  const float* b_delta = (const float*)d_in[8];
  const float* b_gate  = (const float*)d_in[9];

  float* out_p  = (float*)d_out;            // [T,B,D]
  float* hout_p = out_p + (size_t)TBD;      // [T+1,B,D]

  float* vx_ws    = (float*)d_ws;
  float* alpha_ws = vx_ws + (size_t)TBD;
  float* delta_ws = alpha_ws + (size_t)TBD;

  // Phase 1: three fused GEMMs + activations.
  dim3 grid1(MROWS / BM, DIMD / BN);        // (256, 16)
  fused_proj_kernel<<<grid1, 256, 0, stream>>>(
      x, W_x, W_alpha, W_delta, b, b_alpha, b_delta,
      vx_ws, alpha_ws, delta_ws);

  // Phase 2: elementwise recurrence scan.
  const int nthreads = (BATCH * DIMD) / ILP;  // 4096
  scan_kernel<<<nthreads / 256, 256, 0, stream>>>(
      x, h0, r_h, b_gate, vx_ws, alpha_ws, delta_ws, out_p, hout_p);
}